// Linear_66949950210406
// MI455X (gfx1250) — compile-verified
//
#include <hip/hip_runtime.h>
#include <math.h>

// GLN layer for MI455X (gfx1250, wave32).
//   SIZE=1024, INPUT=1024, CTX=512, CMS=4 (16 rows/neuron), BATCH=256
// d_in: 0=logits(1024,256) 1=context_inputs(512,256) 2=targets(256)
//       3=weights(1024,16,1024) 4=context_maps(1024,4,512)
//       5=context_bias(1024,4) 6=bias(1)
// d_out: out(1024,256) ++ new_weights(1024,16,1024)
// d_ws : idx(1024*256 int, 1MB) ++ logitsT(256,1024 f32, 1MB)

typedef float v2f __attribute__((ext_vector_type(2)));
typedef float v8f __attribute__((ext_vector_type(8)));

#define N_SIZE  1024
#define N_INPUT 1024
#define N_CTX   512
#define N_BATCH 256
#define LOGIT_HI 4.59511985013459f   // logit(0.99)
#define LR_F     0.01f
#define WCLIP_F  5.0f

// ---------------------------------------------------------------------------
// Kernel 0: transpose logits (INPUT,BATCH) -> (BATCH,INPUT) for coalesced rows
// ---------------------------------------------------------------------------
__global__ __launch_bounds__(256) void gln_transpose(const float* __restrict__ in,
                                                     float* __restrict__ outT) {
  int t = blockIdx.x * 256 + threadIdx.x;    // 262144 threads
  int b = t >> 10;                           // 0..255
  int k = t & 1023;                          // 0..1023
  outT[t] = in[k * N_BATCH + b];
}

// ---------------------------------------------------------------------------
// Kernel 1: context hashing via V_WMMA_F32_16X16X4_F32.
//   D(4096,256) = context_maps(4096,512) x context_inputs(512,256)
// One wave per 16x16 output tile (= 4 neurons x 4 maps x 16 batch columns).
// C layout: lane<16 holds M=v (VGPR v), lane>=16 holds M=v+8, N=lane&15.
// => each lane owns all 4 context bits of TWO (neuron,batch) entries locally.
// ---------------------------------------------------------------------------
__global__ __launch_bounds__(256) void gln_ctx_hash(const float* __restrict__ cmaps, // (4096,512)
                                                    const float* __restrict__ cin,   // (512,256)
                                                    const float* __restrict__ cbias, // (4096)
                                                    int* __restrict__ idx) {         // (1024,256)
  const int lane  = threadIdx.x & 31;
  const int wave  = threadIdx.x >> 5;
  const int tile  = blockIdx.x * 8 + wave;   // 4096 tiles total
  const int tm    = tile >> 4;               // 0..255 over 4096 rows
  const int tn    = tile & 15;               // 0..15  over 256 cols
  const int m0    = tm << 4;
  const int n0    = tn << 4;
  const int lhalf = lane >> 4;               // 0: K pair {0,1} / M 0..7 ; 1: K {2,3} / M 8..15
  const int l16   = lane & 15;

  const float* arow = cmaps + (size_t)(m0 + l16) * N_CTX + (lhalf << 1);
  const float* bcol = cin + (n0 + l16) + (size_t)(lhalf << 1) * N_BATCH;

  v8f c = {0.f, 0.f, 0.f, 0.f, 0.f, 0.f, 0.f, 0.f};
  for (int k0 = 0; k0 < N_CTX; k0 += 4) {
    v2f a, b;
    a.x = arow[k0];                          // A: M=lane&15, K=k0+2*lhalf+{0,1}
    a.y = arow[k0 + 1];
    b.x = bcol[(size_t)k0 * N_BATCH];        // B: N=lane&15, K=k0+2*lhalf+{0,1}
    b.y = bcol[(size_t)(k0 + 1) * N_BATCH];
    c = __builtin_amdgcn_wmma_f32_16x16x4_f32(false, a, false, b, (short)0, c,
                                              false, false);
  }

  // Pack bits: rows m0+v+8*lhalf; neuron = row>>2, map = row&3.
  int acc0 = 0, acc1 = 0;
#pragma unroll
  for (int v = 0; v < 8; ++v) {
    int R = m0 + v + (lhalf << 3);
    int bit = (c[v] > cbias[R]) ? 1 : 0;
    if (v < 4) acc0 |= bit << (v & 3);
    else       acc1 |= bit << (v & 3);
  }
  const int sA  = (m0 >> 2) + (lhalf << 1);  // first of the two neurons this lane owns
  const int col = n0 + l16;
  idx[(size_t)sA * N_BATCH + col]       = acc0;
  idx[(size_t)(sA + 1) * N_BATCH + col] = acc1;
}

// ---------------------------------------------------------------------------
// Kernel 2: gathered forward. One block per neuron; each wave owns 32 batch
// columns; 32 lanes do a coalesced float4 dot over the selected weight row
// (weights stay L2-resident: 64MB < 192MB L2), then shfl_xor wave reduction.
// ---------------------------------------------------------------------------
__global__ __launch_bounds__(256) void gln_forward(const float* __restrict__ weights, // (1024,16,1024)
                                                   const float* __restrict__ logitsT, // (256,1024)
                                                   const int* __restrict__ idx,       // (1024,256)
                                                   const float* __restrict__ bias,    // (1)
                                                   float* __restrict__ out) {         // (1024,256)
  const int s    = blockIdx.x;
  const int lane = threadIdx.x & 31;
  const int wave = threadIdx.x >> 5;

  for (int b = wave * 32; b < wave * 32 + 32; ++b) {
    const int j = idx[(size_t)s * N_BATCH + b];
    const float4* wrow = (const float4*)(weights + ((size_t)s * 16 + j) * N_INPUT);
    const float4* lrow = (const float4*)(logitsT + (size_t)b * N_INPUT);
    float acc = 0.f;
#pragma unroll
    for (int i = 0; i < 8; ++i) {
      float4 w4 = wrow[lane + i * 32];
      float4 l4 = lrow[lane + i * 32];
      acc += w4.x * l4.x + w4.y * l4.y + w4.z * l4.z + w4.w * l4.w;
    }
#pragma unroll
    for (int off = 16; off > 0; off >>= 1) acc += __shfl_xor(acc, off, 32);
    if (lane == 0) {
      float v = fminf(fmaxf(acc, -LOGIT_HI), LOGIT_HI);
      if (s == 0) v = bias[0];               // bias overrides neuron 0
      out[(size_t)s * N_BATCH + b] = v;
    }
  }
}

// ---------------------------------------------------------------------------
// Kernel 3: weight update + full copy-out (the 128MB HBM-bound pass).
// One block per neuron. winner[j] = last batch b with idx[s,b]==j (scatter-set
// duplicate semantics). Touched rows get clip(w - LR*(sig-t)*logits_col),
// untouched rows are copied verbatim. Fully coalesced (t, t+256, ...).
// ---------------------------------------------------------------------------
__global__ __launch_bounds__(256) void gln_update(const float* __restrict__ weights,
                                                  const float* __restrict__ logitsT,
                                                  const float* __restrict__ targets,
                                                  const int* __restrict__ idx,
                                                  const float* __restrict__ outbuf,
                                                  float* __restrict__ neww) {
  __shared__ int winner[16];
  const int s = blockIdx.x;
  const int t = threadIdx.x;
  if (t < 16) winner[t] = -1;
  __syncthreads();
  atomicMax(&winner[idx[(size_t)s * N_BATCH + t]], t);
  __syncthreads();

  for (int j = 0; j < 16; ++j) {
    const size_t row = ((size_t)s * 16 + j) * N_INPUT;
    const int bstar = winner[j];
    if (bstar < 0) {
#pragma unroll
      for (int i = 0; i < 4; ++i) {
        int k = t + i * 256;
        neww[row + k] = weights[row + k];
      }
    } else {
      const float o    = outbuf[(size_t)s * N_BATCH + bstar];
      const float sig  = 1.f / (1.f + expf(-o));
      const float coef = LR_F * (sig - targets[bstar]);
      const float* lrow = logitsT + (size_t)bstar * N_INPUT;
#pragma unroll
      for (int i = 0; i < 4; ++i) {
        int k = t + i * 256;
        float nv = weights[row + k] - coef * lrow[k];
        neww[row + k] = fminf(fmaxf(nv, -WCLIP_F), WCLIP_F);
      }
    }
  }
}

// ---------------------------------------------------------------------------
extern "C" void kernel_launch(void* const* d_in, const int* in_sizes, int n_in,
                              void* d_out, int out_size, void* d_ws, size_t ws_size,
                              hipStream_t stream) {
  const float* logits  = (const float*)d_in[0];
  const float* cin     = (const float*)d_in[1];
  const float* targets = (const float*)d_in[2];
  const float* weights = (const float*)d_in[3];
  const float* cmaps   = (const float*)d_in[4];
  const float* cbias   = (const float*)d_in[5];
  const float* bias    = (const float*)d_in[6];

  float* out  = (float*)d_out;                       // (1024,256)
  float* neww = out + (size_t)N_SIZE * N_BATCH;      // (1024,16,1024)

  int*   idx     = (int*)d_ws;                                       // 1 MB
  float* logitsT = (float*)((char*)d_ws + (size_t)N_SIZE * N_BATCH * sizeof(int)); // 1 MB

  gln_transpose<<<(N_BATCH * N_INPUT) / 256, 256, 0, stream>>>(logits, logitsT);
  gln_ctx_hash <<<512, 256, 0, stream>>>(cmaps, cin, cbias, idx);    // 4096 waves / tiles
  gln_forward  <<<N_SIZE, 256, 0, stream>>>(weights, logitsT, idx, bias, out);
  gln_update   <<<N_SIZE, 256, 0, stream>>>(weights, logitsT, targets, idx, out, neww);
}